// S6_39144331935826
// MI455X (gfx1250) — compile-verified
//
#include <hip/hip_runtime.h>
#include <math.h>

typedef float v2f __attribute__((ext_vector_type(2)));
typedef float v8f __attribute__((ext_vector_type(8)));

#define BATCH  2
#define SEQ    2048
#define DMODEL 1024
#define DSTATE 16
#define DTRANK 64
#define EPROJ  96              // DT_RANK + 2*D_STATE
#define MROWS  (BATCH * SEQ)   // 4096 flattened (b,l) rows

__device__ __forceinline__ float softplusf(float t) {
    return (t > 20.0f) ? t : log1pf(__expf(t));
}

// ---------------------------------------------------------------------------
// Kernel 1: xp[m, e] = sum_d x[m, d] * x_proj_w[e, d]
// One wave per 16x16 output tile, K swept in steps of 4 via v_wmma_f32_16x16x4_f32.
// Both operands are K-contiguous in memory -> single b64 load per fragment.
// ---------------------------------------------------------------------------
__global__ __launch_bounds__(128) void s6_xproj_gemm(
    const float* __restrict__ x, const float* __restrict__ xw, float* __restrict__ xp)
{
    const int tid   = blockIdx.x * blockDim.x + threadIdx.x;
    const int wave  = tid >> 5;
    const int lane  = tid & 31;
    const int tileN = wave % (EPROJ / 16);          // 6 tiles
    const int tileM = wave / (EPROJ / 16);          // 256 tiles
    const int l15   = lane & 15;
    const int kOff  = (lane >> 4) << 1;             // lanes 0-15: K+{0,1}; 16-31: K+{2,3}
    const int row   = tileM * 16 + l15;             // A row (m)
    const int col   = tileN * 16 + l15;             // B col (e)

    const float* aPtr = x  + (size_t)row * DMODEL + kOff;
    const float* bPtr = xw + (size_t)col * DMODEL + kOff;   // x_proj_w[e, d] row-major

    v8f acc = {};
    #pragma unroll 8
    for (int k0 = 0; k0 < DMODEL; k0 += 4) {
        v2f a = *(const v2f*)(aPtr + k0);
        v2f b = *(const v2f*)(bPtr + k0);
        acc = __builtin_amdgcn_wmma_f32_16x16x4_f32(
            false, a, false, b, (short)0, acc, false, false);
    }

    // D layout: lane&15 = col, vgpr v = row (m & 7), lanes>=16 hold m+8
    const int mBase = tileM * 16 + ((lane >> 4) << 3);
    #pragma unroll
    for (int v = 0; v < 8; ++v)
        xp[(size_t)(mBase + v) * EPROJ + col] = acc[v];
}

// ---------------------------------------------------------------------------
// Kernel 2: delta[m, d] = softplus( sum_r xp[m, r] * dt_proj_w[d, r] + dt_proj_b[d] )
// ---------------------------------------------------------------------------
__global__ __launch_bounds__(128) void s6_delta_gemm(
    const float* __restrict__ xp, const float* __restrict__ dtw,
    const float* __restrict__ dtb, float* __restrict__ delta)
{
    const int tid   = blockIdx.x * blockDim.x + threadIdx.x;
    const int wave  = tid >> 5;
    const int lane  = tid & 31;
    const int tileN = wave % (DMODEL / 16);         // 64 tiles
    const int tileM = wave / (DMODEL / 16);         // 256 tiles
    const int l15   = lane & 15;
    const int kOff  = (lane >> 4) << 1;
    const int row   = tileM * 16 + l15;
    const int col   = tileN * 16 + l15;

    const float* aPtr = xp  + (size_t)row * EPROJ  + kOff;  // first 64 cols of xp
    const float* bPtr = dtw + (size_t)col * DTRANK + kOff;  // dt_proj_w[d, r] row-major

    v8f acc = {};
    #pragma unroll
    for (int k0 = 0; k0 < DTRANK; k0 += 4) {
        v2f a = *(const v2f*)(aPtr + k0);
        v2f b = *(const v2f*)(bPtr + k0);
        acc = __builtin_amdgcn_wmma_f32_16x16x4_f32(
            false, a, false, b, (short)0, acc, false, false);
    }

    const int mBase = tileM * 16 + ((lane >> 4) << 3);
    const float bias = dtb[col];
    #pragma unroll
    for (int v = 0; v < 8; ++v)
        delta[(size_t)(mBase + v) * DMODEL + col] = softplusf(acc[v] + bias);
}

// ---------------------------------------------------------------------------
// Kernel 3: selective scan. One wave owns 2 d-channels x 16 state dims.
//   lane = 16*dSub + n.  h kept in a register for all 2048 steps.
//   h = exp(delta*A[d,n])*h + delta*B[l,n]*x ;  y = sum_n h*C[l,n] + x*D[d]
// Cross-lane sum over n via 16-wide shfl_xor (ds lane permutes, no LDS).
// Loads for step l+1 issued before the step-l reduction (software pipeline).
// ---------------------------------------------------------------------------
__global__ __launch_bounds__(256) void s6_scan(
    const float* __restrict__ x, const float* __restrict__ xp,
    const float* __restrict__ delta, const float* __restrict__ A_log,
    const float* __restrict__ Dvec, float* __restrict__ y)
{
    const int tid  = blockIdx.x * blockDim.x + threadIdx.x;
    const int wave = tid >> 5;
    const int lane = tid & 31;
    const int n    = lane & 15;
    const int dSub = lane >> 4;
    const int b    = wave / (DMODEL / 2);
    const int d    = (wave % (DMODEL / 2)) * 2 + dSub;

    const float A_dn = -__expf(A_log[d * DSTATE + n]);
    const float Dd   = Dvec[d];

    const float* xB   = x     + (size_t)b * SEQ * DMODEL + d;
    const float* dlB  = delta + (size_t)b * SEQ * DMODEL + d;
    const float* xpB  = xp    + (size_t)b * SEQ * EPROJ;
    float*       yB   = y     + (size_t)b * SEQ * DMODEL + d;

    float h = 0.0f;

    // prologue: stage l = 0
    float dl = dlB[0];
    float xv = xB[0];
    float Bn = xpB[DTRANK + n];
    float Cn = xpB[DTRANK + DSTATE + n];

    for (int l = 0; l < SEQ; ++l) {
        // issue next step's loads early (hide latency behind exp + reduction)
        float dl_n = 0.f, xv_n = 0.f, Bn_n = 0.f, Cn_n = 0.f;
        if (l + 1 < SEQ) {
            const size_t r = (size_t)(l + 1);
            dl_n = dlB[r * DMODEL];
            xv_n = xB[r * DMODEL];
            Bn_n = xpB[r * EPROJ + DTRANK + n];
            Cn_n = xpB[r * EPROJ + DTRANK + DSTATE + n];
        }

        const float dA = __expf(dl * A_dn);
        h = fmaf(dA, h, dl * Bn * xv);

        float yp = h * Cn;
        yp += __shfl_xor(yp, 1, 16);
        yp += __shfl_xor(yp, 2, 16);
        yp += __shfl_xor(yp, 4, 16);
        yp += __shfl_xor(yp, 8, 16);

        if (n == 0)
            yB[(size_t)l * DMODEL] = fmaf(xv, Dd, yp);

        dl = dl_n; xv = xv_n; Bn = Bn_n; Cn = Cn_n;
    }
}

// ---------------------------------------------------------------------------
extern "C" void kernel_launch(void* const* d_in, const int* in_sizes, int n_in,
                              void* d_out, int out_size, void* d_ws, size_t ws_size,
                              hipStream_t stream) {
    const float* x     = (const float*)d_in[0];   // (2, 2048, 1024)
    const float* A_log = (const float*)d_in[1];   // (1024, 16)
    const float* Dv    = (const float*)d_in[2];   // (1024,)
    const float* xw    = (const float*)d_in[3];   // (96, 1024)
    const float* dtw   = (const float*)d_in[4];   // (1024, 64)
    const float* dtb   = (const float*)d_in[5];   // (1024,)
    float* yOut        = (float*)d_out;

    float* xp    = (float*)d_ws;                                       // 4096*96 f32
    float* delta = (float*)((char*)d_ws +
                            (size_t)MROWS * EPROJ * sizeof(float));    // 4096*1024 f32

    // GEMM1: 256*6 = 1536 waves, 4 waves/block
    s6_xproj_gemm<<<(MROWS / 16) * (EPROJ / 16) / 4, 128, 0, stream>>>(x, xw, xp);
    // GEMM2: 256*64 = 16384 waves, 4 waves/block
    s6_delta_gemm<<<(MROWS / 16) * (DMODEL / 16) / 4, 128, 0, stream>>>(xp, dtw, dtb, delta);
    // Scan: 2*512 = 1024 waves, 8 waves/block
    s6_scan<<<(BATCH * (DMODEL / 2)) / 8, 256, 0, stream>>>(x, xp, delta, A_log, Dv, yOut);
}